// GoMHCResidual_42649025249998
// MI455X (gfx1250) — compile-verified
//
#include <hip/hip_runtime.h>

#define NSTREAMS 8
#define DMODEL   2048
#define NS       16
#define NSKEW    120
#define KCHUNK   64
#define MTILE    32
#define LDA      68   // padded LDS row stride (floats): 68 mod 64 = 4 -> conflict-free column reads

typedef __attribute__((ext_vector_type(2))) float v2f;
typedef __attribute__((ext_vector_type(8))) float v8f;

// ---------------------------------------------------------------------------
// Kernel 1: fused  z = mean_streams(streams) @ W^T + b   via fp32 WMMA
//   streams: [T, 8, 2048] fp32   W: [120, 2048] fp32   b: [120]
//   z (workspace): [T, 120] fp32
// Each WG: 32 tokens (M-tile). 8 waves: wave w -> row-group g=w&1 (16 rows),
// col-group c=w>>1 (32 of 128 padded N columns -> 2 v8f accumulators).
// ---------------------------------------------------------------------------
__global__ __launch_bounds__(256) void fused_mean_gemm_kernel(
    const float* __restrict__ streams, const float* __restrict__ W,
    const float* __restrict__ bias, float* __restrict__ z, int T)
{
    __shared__ float xs[MTILE * LDA];   // x_agg chunk [32][64] (padded)
    __shared__ float wl[128 * LDA];     // W^T chunk  [128][64] (rows 120..127 zero)

    const int tid  = threadIdx.x;
    const int lane = tid & 31;
    const int wave = tid >> 5;
    const int half = lane >> 4;     // 0: lanes 0-15, 1: lanes 16-31
    const int l16  = lane & 15;
    const int g    = wave & 1;      // M row-group within tile
    const int c    = wave >> 1;     // N col-group (32 cols)
    const int t0   = blockIdx.x * MTILE;

    // bias preload for the two N-tiles this wave owns (N = column index = lane%16 based)
    float bv[2];
#pragma unroll
    for (int nt = 0; nt < 2; ++nt) {
        int n = (2 * c + nt) * 16 + l16;
        bv[nt] = (n < NSKEW) ? bias[n] : 0.0f;
    }

    v8f acc0 = {};
    v8f acc1 = {};

    const float4* s4 = (const float4*)streams;
    const float4* w4 = (const float4*)W;

    for (int k0 = 0; k0 < DMODEL; k0 += KCHUNK) {
        __syncthreads();
        // ---- stage x_agg chunk: 32 tokens x 16 float4 = 512 (t,k4) pairs ----
#pragma unroll
        for (int r = 0; r < 2; ++r) {
            int p  = tid + 256 * r;
            int t  = p >> 4;
            int k4 = p & 15;
            float4 sum = make_float4(0.f, 0.f, 0.f, 0.f);
#pragma unroll
            for (int s = 0; s < NSTREAMS; ++s) {
                float4 v = s4[(size_t)((t0 + t) * NSTREAMS + s) * (DMODEL / 4) + (k0 >> 2) + k4];
                sum.x += v.x; sum.y += v.y; sum.z += v.z; sum.w += v.w;
            }
            float* dst = &xs[t * LDA + k4 * 4];
            dst[0] = sum.x * 0.125f; dst[1] = sum.y * 0.125f;
            dst[2] = sum.z * 0.125f; dst[3] = sum.w * 0.125f;
        }
        // ---- stage W chunk (B matrix = W^T): 128 x 16 float4 pairs ----
#pragma unroll
        for (int r = 0; r < 8; ++r) {
            int p  = tid + 256 * r;
            int n  = p >> 4;
            int k4 = p & 15;
            float4 v = make_float4(0.f, 0.f, 0.f, 0.f);
            if (n < NSKEW) v = w4[(size_t)n * (DMODEL / 4) + (k0 >> 2) + k4];
            float* dst = &wl[n * LDA + k4 * 4];
            dst[0] = v.x; dst[1] = v.y; dst[2] = v.z; dst[3] = v.w;
        }
        __syncthreads();

        // ---- WMMA over the chunk: 16 k-steps of K=4 ----
#pragma unroll
        for (int ks = 0; ks < KCHUNK / 4; ++ks) {
            int kk = ks * 4 + half * 2;   // A/B frag: lanes 0-15 hold K=kk,kk+1; 16-31 hold K+2
            // A fragment: 16x4 fp32, lane row = 16*g + l16
            v2f a = *(const v2f*)&xs[(16 * g + l16) * LDA + kk];
            // B fragments: 4x16 fp32, B[k][n] = W[n][k]
            v2f b0 = *(const v2f*)&wl[((2 * c + 0) * 16 + l16) * LDA + kk];
            acc0 = __builtin_amdgcn_wmma_f32_16x16x4_f32(false, a, false, b0,
                                                         (short)0, acc0, false, false);
            v2f b1 = *(const v2f*)&wl[((2 * c + 1) * 16 + l16) * LDA + kk];
            acc1 = __builtin_amdgcn_wmma_f32_16x16x4_f32(false, a, false, b1,
                                                         (short)0, acc1, false, false);
        }
    }

    // ---- epilogue: C VGPR layout -> z[row][n] = acc + bias, n < 120 only ----
#pragma unroll
    for (int r = 0; r < 8; ++r) {
        int m   = half ? (r + 8) : r;       // M index per documented C/D layout
        int row = t0 + 16 * g + m;
        int n0  = (2 * c + 0) * 16 + l16;
        if (n0 < NSKEW) z[(size_t)row * NSKEW + n0] = acc0[r] + bv[0];
        int n1  = (2 * c + 1) * 16 + l16;
        if (n1 < NSKEW) z[(size_t)row * NSKEW + n1] = acc1[r] + bv[1];
    }
}

// ---------------------------------------------------------------------------
// Kernel 2: per-token Cayley transform + 2x2 block Frobenius norms.
// 16-lane segment per token (2 tokens/wave). Lane i owns row i of
// M = I+A and R = I-A in registers. Gauss-Jordan (no pivoting; symmetric
// part of I+A is I, SPD => stable). After reduction R = Q = (I+A)^{-1}(I-A).
// ---------------------------------------------------------------------------
__global__ __launch_bounds__(128) void cayley_block_norm_kernel(
    const float* __restrict__ z, float* __restrict__ out, int T)
{
    const int lane  = threadIdx.x & 31;
    const int wave  = threadIdx.x >> 5;
    const int seg   = lane >> 4;
    const int i     = lane & 15;
    const int token = blockIdx.x * 8 + wave * 2 + seg;
    if (token >= T) return;   // uniform within each 16-lane segment

    const float* zp = z + (size_t)token * NSKEW;

    float M[16], R[16];
#pragma unroll
    for (int j = 0; j < 16; ++j) {
        float a;
        if (j == i)      a = 0.0f;
        else if (i < j)  a =  zp[i * (31 - i) / 2 + (j - i - 1)];
        else             a = -zp[j * (31 - j) / 2 + (i - j - 1)];
        float d = (j == i) ? 1.0f : 0.0f;
        M[j] = d + a;
        R[j] = d - a;
    }

    // Gauss-Jordan: reduce M -> I, R -> Q. k static via full unroll.
#pragma unroll
    for (int k = 0; k < 16; ++k) {
        float pk[16], pr[16];
#pragma unroll
        for (int j = 0; j < 16; ++j) {
            pk[j] = __shfl(M[j], k, 16);
            pr[j] = __shfl(R[j], k, 16);
        }
        float inv = 1.0f / pk[k];
        if (i == k) {
#pragma unroll
            for (int j = 0; j < 16; ++j) { M[j] *= inv; R[j] *= inv; }
        } else {
            float f = M[k] * inv;
#pragma unroll
            for (int j = 0; j < 16; ++j) { M[j] -= f * pk[j]; R[j] -= f * pr[j]; }
        }
    }

    // H[n][m] = (sum over 2x2 block of Q^2) / 2 ; lane i holds Q row i.
    float p[8];
#pragma unroll
    for (int m = 0; m < 8; ++m)
        p[m] = R[2 * m] * R[2 * m] + R[2 * m + 1] * R[2 * m + 1];
#pragma unroll
    for (int m = 0; m < 8; ++m) {
        float q = p[m] + __shfl_xor(p[m], 1, 16);   // combine rows 2n, 2n+1
        if ((i & 1) == 0)
            out[(size_t)token * 64 + (i >> 1) * 8 + m] = q * 0.5f;
    }
}

// ---------------------------------------------------------------------------
extern "C" void kernel_launch(void* const* d_in, const int* in_sizes, int n_in,
                              void* d_out, int out_size, void* d_ws, size_t ws_size,
                              hipStream_t stream) {
    const float* streams = (const float*)d_in[0];   // [B,L,8,2048] fp32
    const float* W       = (const float*)d_in[1];   // [120,2048]   fp32
    const float* bias    = (const float*)d_in[2];   // [120]        fp32
    float*       out     = (float*)d_out;           // [B,L,8,8]    fp32

    const int T = in_sizes[0] / (NSTREAMS * DMODEL);   // B*L = 8192
    float* z = (float*)d_ws;                           // [T,120] fp32 scratch

    fused_mean_gemm_kernel<<<T / MTILE, 256, 0, stream>>>(streams, W, bias, z, T);
    cayley_block_norm_kernel<<<(T + 7) / 8, 128, 0, stream>>>(z, out, T);
}